// Head_48661979464193
// MI455X (gfx1250) — compile-verified
//
#include <hip/hip_runtime.h>

#define B_DIM 8
#define T_DIM 2048
#define C_DIM 1024
#define H_DIM 64
#define WPB   8            // waves per block (wave32)
#define TPB   (WPB * 32)
#define KC    32           // C-columns staged per LDS chunk
#define NCH   (C_DIM / KC) // 32 chunks

typedef float v2f __attribute__((ext_vector_type(2)));
typedef float v8f __attribute__((ext_vector_type(8)));

// D = A(16x4) * B(4x16) + C, full fp32 (codegen-verified round 0).
__device__ __forceinline__ v8f wmma_f32(v2f a, v2f b, v8f c) {
  return __builtin_amdgcn_wmma_f32_16x16x4_f32(false, a, false, b, (short)0, c,
                                               false, false);
}

// Async-stage one KC-column chunk of Wk and Wv into LDS (ASYNCcnt path).
// 64 rows x KC=32 cols = 2048 floats per matrix; thread t moves 8 floats of
// each matrix as two b128 transfers (INST_OFFSET applies to both LDS+global).
__device__ __forceinline__ void stage_chunk(const float* __restrict__ Wk,
                                            const float* __restrict__ Wv,
                                            float* wk_lds, float* wv_lds,
                                            int cBase, int t) {
  const int f0  = t * 8;
  const int row = f0 >> 5;   // / KC
  const int col = f0 & 31;   // % KC
  const float* gk = Wk + row * C_DIM + cBase + col;
  const float* gv = Wv + row * C_DIM + cBase + col;
  unsigned lk = (unsigned)(size_t)(wk_lds + f0);
  unsigned lv = (unsigned)(size_t)(wv_lds + f0);
  asm volatile("global_load_async_to_lds_b128 %0, %1, off"
               :: "v"(lk), "v"(gk) : "memory");
  asm volatile("global_load_async_to_lds_b128 %0, %1, off offset:16"
               :: "v"(lk), "v"(gk) : "memory");
  asm volatile("global_load_async_to_lds_b128 %0, %1, off"
               :: "v"(lv), "v"(gv) : "memory");
  asm volatile("global_load_async_to_lds_b128 %0, %1, off offset:16"
               :: "v"(lv), "v"(gv) : "memory");
}

// ---------------------------------------------------------------------------
// Kernel 1: K = x @ Wk^T (row-major [B*T,64]); V = x @ Wv^T stored transposed
// per 16-key tile: Vt[tile][h][key16]. W staged via async-to-LDS, double buf.
// ---------------------------------------------------------------------------
__global__ __launch_bounds__(TPB) void head_proj_kernel(
    const float* __restrict__ x, const float* __restrict__ Wk,
    const float* __restrict__ Wv, float* __restrict__ Kout,
    float* __restrict__ Vt) {
  __shared__ float wk_lds[2][64 * KC];
  __shared__ float wv_lds[2][64 * KC];
  const int t = threadIdx.x;
  const int lane = t & 31;
  const int wave = t >> 5;
  const int tile = blockIdx.x * WPB + wave;        // 16-row tile of [B*T]
  const long rowBase = (long)tile * 16;
  const int g = lane >> 4;
  const int n = lane & 15;

  v8f accK[4] = {};
  v8f accV[4] = {};
  const float* xrow = x + (rowBase + n) * C_DIM;   // A: M = lane&15

  stage_chunk(Wk, Wv, wk_lds[0], wv_lds[0], 0, t);

  for (int c = 0; c < NCH; ++c) {
    asm volatile("s_wait_asynccnt 0x0" ::: "memory");
    __syncthreads();                               // chunk c resident in LDS
    if (c + 1 < NCH)
      stage_chunk(Wk, Wv, wk_lds[(c + 1) & 1], wv_lds[(c + 1) & 1],
                  (c + 1) * KC, t);                // DMA next chunk in bg

    const float* wk = wk_lds[c & 1];
    const float* wv = wv_lds[c & 1];
    const float* xc = xrow + c * KC;

    // register double-buffer: preload step 0
    v2f a_c = *(const v2f*)(xc + 2 * g);
    v2f bk_c[4], bv_c[4];
#pragma unroll
    for (int c2 = 0; c2 < 4; ++c2) {
      bk_c[c2] = *(const v2f*)(wk + (c2 * 16 + n) * KC + 2 * g);
      bv_c[c2] = *(const v2f*)(wv + (c2 * 16 + n) * KC + 2 * g);
    }
#pragma unroll
    for (int s = 0; s < KC / 4; ++s) {
      v2f a_n = a_c;
      v2f bk_n[4], bv_n[4];
      if (s + 1 < KC / 4) {
        const int ko = 4 * (s + 1) + 2 * g;
        a_n = *(const v2f*)(xc + ko);
#pragma unroll
        for (int c2 = 0; c2 < 4; ++c2) {
          bk_n[c2] = *(const v2f*)(wk + (c2 * 16 + n) * KC + ko);
          bv_n[c2] = *(const v2f*)(wv + (c2 * 16 + n) * KC + ko);
        }
      } else {
#pragma unroll
        for (int c2 = 0; c2 < 4; ++c2) { bk_n[c2] = bk_c[c2]; bv_n[c2] = bv_c[c2]; }
      }
#pragma unroll
      for (int c2 = 0; c2 < 4; ++c2) {
        accK[c2] = wmma_f32(a_c, bk_c[c2], accK[c2]);
        accV[c2] = wmma_f32(a_c, bv_c[c2], accV[c2]);
      }
      a_c = a_n;
#pragma unroll
      for (int c2 = 0; c2 < 4; ++c2) { bk_c[c2] = bk_n[c2]; bv_c[c2] = bv_n[c2]; }
    }
  }

  // C/D layout: VGPR r holds row (r + 8*g), col = (c2*16 + n)
#pragma unroll
  for (int c2 = 0; c2 < 4; ++c2) {
#pragma unroll
    for (int r = 0; r < 8; ++r) {
      const int row = r + 8 * g;                   // row within tile
      Kout[(rowBase + row) * H_DIM + c2 * 16 + n] = accK[c2][r];
      // Vt[tile][h][key]
      Vt[((long)tile * H_DIM + c2 * 16 + n) * 16 + row] = accV[c2][r];
    }
  }
}

// ---------------------------------------------------------------------------
// Kernel 2: flash-style causal attention, q == k (reference "bug").
// One wave owns one 16-query tile; online softmax over key tiles st<=qt.
// ---------------------------------------------------------------------------
__global__ __launch_bounds__(TPB) void head_attn_kernel(
    const float* __restrict__ Kin, const float* __restrict__ Vt,
    float* __restrict__ out) {
  __shared__ float p_stage[WPB][16 * 16];          // per-wave P staging
  const int lane = threadIdx.x & 31;
  const int wave = threadIdx.x >> 5;
  const int tile = blockIdx.x * WPB + wave;        // 1024 tiles total
  const int b  = tile >> 7;                        // T/16 = 128 tiles/batch
  const int qt = tile & 127;
  const int g = lane >> 4;
  const int n = lane & 15;
  const long qBase = (long)qt * 16;

  const float* Kb = Kin + (long)b * T_DIM * H_DIM;
  float* pw = &p_stage[wave][0];

  // Query-tile A fragments are st-invariant: load once (16 steps of K=4).
  const float* aRow = Kb + (qBase + n) * H_DIM;
  v2f aq[16];
#pragma unroll
  for (int s = 0; s < 16; ++s) aq[s] = *(const v2f*)(aRow + 4 * s + 2 * g);

  v8f O[4] = {};
  float Mrow[8], Lrow[8];
#pragma unroll
  for (int r = 0; r < 8; ++r) { Mrow[r] = -__builtin_inff(); Lrow[r] = 0.f; }

  for (int st = 0; st <= qt; ++st) {
    const long sBase = (long)st * 16;
    const float* kS  = Kb + (sBase + n) * H_DIM;       // key row sBase+n
    const float* vtS = Vt + ((long)(b * 128 + st) * H_DIM) * 16;

    if (st < qt) {                                     // prefetch next tile
      __builtin_prefetch(Kb + (sBase + 16 + n) * H_DIM, 0, 1);
      __builtin_prefetch(vtS + H_DIM * 16 + lane * 32, 0, 1);
    }

    // S = Kq(16x64) * Ks^T(64x16), B fragments double-buffered
    v8f S = {};
    v2f bt_c = *(const v2f*)(kS + 2 * g);
#pragma unroll
    for (int s = 0; s < 16; ++s) {
      v2f bt_n = bt_c;
      if (s + 1 < 16) bt_n = *(const v2f*)(kS + 4 * (s + 1) + 2 * g);
      S = wmma_f32(aq[s], bt_c, S);
      bt_c = bt_n;
    }

    const bool diag = (st == qt);
#pragma unroll
    for (int r = 0; r < 8; ++r) {
      const int row = r + 8 * g;                   // query row within tile
      float s = S[r] * 0.125f;                     // 1/sqrt(64)
      if (diag && n > row) s = -__builtin_inff();  // causal mask
      float v = s;                                 // row-max over 16 lanes
      v = fmaxf(v, __shfl_xor(v, 1));
      v = fmaxf(v, __shfl_xor(v, 2));
      v = fmaxf(v, __shfl_xor(v, 4));
      v = fmaxf(v, __shfl_xor(v, 8));
      const float mnew  = fmaxf(Mrow[r], v);
      const float alpha = __expf(Mrow[r] - mnew);
      Mrow[r] = mnew;
      const float p = __expf(s - mnew);
      float sum = p;
      sum += __shfl_xor(sum, 1);
      sum += __shfl_xor(sum, 2);
      sum += __shfl_xor(sum, 4);
      sum += __shfl_xor(sum, 8);
      Lrow[r] = Lrow[r] * alpha + sum;
#pragma unroll
      for (int c2 = 0; c2 < 4; ++c2) O[c2][r] *= alpha;
      pw[row * 16 + n] = p;                        // stage P for A reload
    }

    // O += P(16x16) * Vtile(16x64); per-wave LDS RAW is in-order on CDNA5.
    // B fragment from Vt is contiguous: Vt[tile][h = c2*16+n][ka..ka+1].
    v2f ap_c = *(const v2f*)(pw + n * 16 + 2 * g);
#pragma unroll
    for (int s = 0; s < 4; ++s) {
      const int ka = 4 * s + 2 * g;
      v2f ap_n = ap_c;
      if (s + 1 < 4) ap_n = *(const v2f*)(pw + n * 16 + 4 * (s + 1) + 2 * g);
      v2f bv_c[4];
#pragma unroll
      for (int c2 = 0; c2 < 4; ++c2)
        bv_c[c2] = *(const v2f*)(vtS + (c2 * 16 + n) * 16 + ka);
#pragma unroll
      for (int c2 = 0; c2 < 4; ++c2)
        O[c2] = wmma_f32(ap_c, bv_c[c2], O[c2]);
      ap_c = ap_n;
    }
  }

  float invL[8];
#pragma unroll
  for (int r = 0; r < 8; ++r) invL[r] = 1.0f / Lrow[r];
#pragma unroll
  for (int c2 = 0; c2 < 4; ++c2) {
#pragma unroll
    for (int r = 0; r < 8; ++r) {
      out[((long)b * T_DIM + qBase + r + 8 * g) * H_DIM + c2 * 16 + n] =
          O[c2][r] * invL[r];
    }
  }
}

extern "C" void kernel_launch(void* const* d_in, const int* in_sizes, int n_in,
                              void* d_out, int out_size, void* d_ws,
                              size_t ws_size, hipStream_t stream) {
  const float* x  = (const float*)d_in[0];
  const float* Wk = (const float*)d_in[1];
  const float* Wv = (const float*)d_in[2];
  float* out = (float*)d_out;

  float* Kbuf  = (float*)d_ws;                              // 4 MB
  float* VtBuf = Kbuf + (size_t)B_DIM * T_DIM * H_DIM;      // 4 MB

  const int tiles = (B_DIM * T_DIM) / 16;                   // 1024
  dim3 grid(tiles / WPB);                                   // 128 blocks
  dim3 block(TPB);                                          // 256 = 8 wave32

  head_proj_kernel<<<grid, block, 0, stream>>>(x, Wk, Wv, Kbuf, VtBuf);
  head_attn_kernel<<<grid, block, 0, stream>>>(Kbuf, VtBuf, out);
}